// GraphSAGE_81681688035405
// MI455X (gfx1250) — compile-verified
//
#include <hip/hip_runtime.h>

typedef __attribute__((ext_vector_type(2))) float v2f;
typedef __attribute__((ext_vector_type(8))) float v8f;

#define NN   50000
#define EE   800000
#define DF   128
#define DOUT 2
#define LDSP (DF + 4)   // padded row stride: conflict-free + 8B-aligned v2f reads

// ---------------- utility kernels ----------------

__global__ __launch_bounds__(256) void zero_kernel(float* __restrict__ p, long n) {
    long i = (long)blockIdx.x * blockDim.x + threadIdx.x;
    long stride = (long)gridDim.x * blockDim.x;
    for (; i < n; i += stride) p[i] = 0.0f;
}

__global__ __launch_bounds__(256) void degree_kernel(const int* __restrict__ dst,
                                                     float* __restrict__ deg) {
    int e = blockIdx.x * blockDim.x + threadIdx.x;
    if (e < EE) atomicAdd(&deg[dst[e]], 1.0f);
}

__global__ __launch_bounds__(256) void recip_kernel(float* __restrict__ deg) {
    int i = blockIdx.x * blockDim.x + threadIdx.x;
    if (i < NN) deg[i] = 1.0f / fmaxf(deg[i], 1.0f);
}

// One wave per edge: lane handles 4 consecutive floats of the 128-wide row.
__global__ __launch_bounds__(256) void scatter_kernel(const float* __restrict__ feat,
                                                      const int* __restrict__ src,
                                                      const int* __restrict__ dst,
                                                      float* __restrict__ agg) {
    long t = (long)blockIdx.x * blockDim.x + threadIdx.x;
    long e = t >> 5;
    if (e >= EE) return;
    int c = (int)(t & 31) * 4;
    const float4 v = *(const float4*)(feat + (long)src[e] * DF + c);
    float* a = agg + (long)dst[e] * DF + c;
    atomicAdd(a + 0, v.x);
    atomicAdd(a + 1, v.y);
    atomicAdd(a + 2, v.z);
    atomicAdd(a + 3, v.w);
}

// ---------------- layer 1: h = relu(mean @ W1l + x @ W1r + b1) ----------------
// Block = 32 output rows, 128 threads (4 waves). Each wave owns 2 col-tiles and
// computes TWO 16x16 accumulators per tile, reusing each B fragment across both
// row-tiles. K loop unrolled x8 to bound VGPR pressure.

template <bool TWO>
__device__ __forceinline__ void gemm1_tile(const float (*Am)[LDSP], const float (*Ax)[LDSP],
                                           const float* __restrict__ Wl,
                                           const float* __restrict__ Wr,
                                           const float* __restrict__ bias,
                                           float* __restrict__ out,
                                           int m0, int cb, int hf, int nl) {
    v8f acc0 = {};
    v8f acc1 = {};
    // fold lane-constant offsets into base pointers -> immediate-offset loads
    const float* wl  = Wl + (long)(2 * hf) * DF + cb + nl;
    const float* wr  = Wr + (long)(2 * hf) * DF + cb + nl;
    const float* am0 = &Am[nl][2 * hf];
    const float* ax0 = &Ax[nl][2 * hf];
    const float* am1 = &Am[16 + nl][2 * hf];
    const float* ax1 = &Ax[16 + nl][2 * hf];

#pragma unroll 8
    for (int k0 = 0; k0 < DF; k0 += 4) {
        v2f bl, br;
        bl.x = wl[(long)k0 * DF]; bl.y = wl[(long)(k0 + 1) * DF];
        br.x = wr[(long)k0 * DF]; br.y = wr[(long)(k0 + 1) * DF];
        v2f a_m0 = *(const v2f*)(am0 + k0);
        v2f a_x0 = *(const v2f*)(ax0 + k0);
        acc0 = __builtin_amdgcn_wmma_f32_16x16x4_f32(false, a_m0, false, bl,
                                                     (short)0, acc0, false, false);
        acc0 = __builtin_amdgcn_wmma_f32_16x16x4_f32(false, a_x0, false, br,
                                                     (short)0, acc0, false, false);
        if (TWO) {
            v2f a_m1 = *(const v2f*)(am1 + k0);
            v2f a_x1 = *(const v2f*)(ax1 + k0);
            acc1 = __builtin_amdgcn_wmma_f32_16x16x4_f32(false, a_m1, false, bl,
                                                         (short)0, acc1, false, false);
            acc1 = __builtin_amdgcn_wmma_f32_16x16x4_f32(false, a_x1, false, br,
                                                         (short)0, acc1, false, false);
        }
    }
    const float bv = bias[cb + nl];
#pragma unroll
    for (int v = 0; v < 8; ++v) {
        // C/D layout: vgpr v, lane -> row (v + 8*half), col nl
        out[(long)(m0 + v + 8 * hf) * DF + cb + nl] = fmaxf(acc0[v] + bv, 0.0f);
        if (TWO)
            out[(long)(m0 + 16 + v + 8 * hf) * DF + cb + nl] = fmaxf(acc1[v] + bv, 0.0f);
    }
}

__global__ __launch_bounds__(128) void gemm1_kernel(const float* __restrict__ x,
                                                    const float* __restrict__ agg,
                                                    const float* __restrict__ dinv,
                                                    const float* __restrict__ Wl,
                                                    const float* __restrict__ Wr,
                                                    const float* __restrict__ bias,
                                                    float* __restrict__ out) {
    __shared__ float Am[32][LDSP];   // mean rows
    __shared__ float Ax[32][LDSP];   // x rows
    const int m0   = blockIdx.x * 32;
    const int tid  = threadIdx.x;
    const int nrow = (NN - m0 < 32) ? (NN - m0) : 32;   // 32 or 16 (NN % 16 == 0)

    // Cooperative, coalesced load of the nrow x 128 A-panels (float4 per thread).
    for (int i = tid; i < nrow * (DF / 4); i += 128) {
        int r = i >> 5;
        int c = (i & 31) * 4;
        long row = m0 + r;
        float di = dinv[row];
        float4 xa = *(const float4*)(x   + row * DF + c);
        float4 ag = *(const float4*)(agg + row * DF + c);
        Ax[r][c + 0] = xa.x; Ax[r][c + 1] = xa.y; Ax[r][c + 2] = xa.z; Ax[r][c + 3] = xa.w;
        Am[r][c + 0] = ag.x * di; Am[r][c + 1] = ag.y * di;
        Am[r][c + 2] = ag.z * di; Am[r][c + 3] = ag.w * di;
    }
    __syncthreads();

    const int wave = tid >> 5;
    const int lane = tid & 31;
    const int hf   = lane >> 4;
    const int nl   = lane & 15;

    if (nrow == 32) {   // uniform (blockIdx-derived) -> scalar branch, EXEC stays full
        for (int ct = wave; ct < DF / 16; ct += 4)
            gemm1_tile<true>(Am, Ax, Wl, Wr, bias, out, m0, ct * 16, hf, nl);
    } else {
        for (int ct = wave; ct < DF / 16; ct += 4)
            gemm1_tile<false>(Am, Ax, Wl, Wr, bias, out, m0, ct * 16, hf, nl);
    }
}

// ---------------- layer 2: out = mean2 @ W2l + h @ W2r + b2 (DOUT=2) ----------
// One wave per 16-row tile; W2 (128x2) zero-padded to 16 cols in registers so
// EXEC stays all-ones around WMMAs; store masked to the 2 real columns.
__global__ __launch_bounds__(32) void gemm2_kernel(const float* __restrict__ h,
                                                   const float* __restrict__ agg,
                                                   const float* __restrict__ dinv,
                                                   const float* __restrict__ Wl,
                                                   const float* __restrict__ Wr,
                                                   const float* __restrict__ bias,
                                                   float* __restrict__ out) {
    __shared__ float Am[16][LDSP];
    __shared__ float Ah[16][LDSP];
    const int m0  = blockIdx.x * 16;
    const int tid = threadIdx.x;   // 0..31

    for (int i = tid; i < 16 * (DF / 4); i += 32) {
        int r = i >> 5;
        int c = (i & 31) * 4;
        long row = m0 + r;
        float di = dinv[row];
        float4 hv = *(const float4*)(h   + row * DF + c);
        float4 ag = *(const float4*)(agg + row * DF + c);
        Ah[r][c + 0] = hv.x; Ah[r][c + 1] = hv.y; Ah[r][c + 2] = hv.z; Ah[r][c + 3] = hv.w;
        Am[r][c + 0] = ag.x * di; Am[r][c + 1] = ag.y * di;
        Am[r][c + 2] = ag.z * di; Am[r][c + 3] = ag.w * di;
    }
    __syncthreads();

    const int hf = tid >> 4;
    const int nl = tid & 15;
    const bool act = (nl < DOUT);

    const float* amp = &Am[nl][2 * hf];
    const float* ahp = &Ah[nl][2 * hf];
    const float* wl  = Wl + (long)(2 * hf) * DOUT + (act ? nl : 0);
    const float* wr  = Wr + (long)(2 * hf) * DOUT + (act ? nl : 0);

    v8f acc = {};
#pragma unroll 8
    for (int k0 = 0; k0 < DF; k0 += 4) {
        v2f am, ah, bl, br;
        am = *(const v2f*)(amp + k0);
        ah = *(const v2f*)(ahp + k0);
        bl.x = act ? wl[(long)k0 * DOUT]       : 0.0f;
        bl.y = act ? wl[(long)(k0 + 1) * DOUT] : 0.0f;
        br.x = act ? wr[(long)k0 * DOUT]       : 0.0f;
        br.y = act ? wr[(long)(k0 + 1) * DOUT] : 0.0f;
        acc = __builtin_amdgcn_wmma_f32_16x16x4_f32(false, am, false, bl,
                                                    (short)0, acc, false, false);
        acc = __builtin_amdgcn_wmma_f32_16x16x4_f32(false, ah, false, br,
                                                    (short)0, acc, false, false);
    }
    if (act) {
        const float bv = bias[nl];
#pragma unroll
        for (int v = 0; v < 8; ++v)
            out[(long)(m0 + v + 8 * hf) * DOUT + nl] = acc[v] + bv;
    }
}

// ---------------- launcher ----------------

extern "C" void kernel_launch(void* const* d_in, const int* in_sizes, int n_in,
                              void* d_out, int out_size, void* d_ws, size_t ws_size,
                              hipStream_t stream) {
    const float* x   = (const float*)d_in[0];
    const int*   ei  = (const int*)  d_in[1];   // [2, E] flat: src then dst
    const float* W1l = (const float*)d_in[2];
    const float* b1  = (const float*)d_in[3];
    const float* W1r = (const float*)d_in[4];
    const float* W2l = (const float*)d_in[5];
    const float* b2  = (const float*)d_in[6];
    const float* W2r = (const float*)d_in[7];
    float* out = (float*)d_out;

    const int* src = ei;
    const int* dst = ei + EE;

    float* agg  = (float*)d_ws;                  // N*DF  (25.6 MB, L2-resident)
    float* hbuf = agg + (size_t)NN * DF;         // N*DF
    float* dinv = hbuf + (size_t)NN * DF;        // N (deg, then deg_inv in place)

    const int scatter_blocks = (int)(((long)EE * 32 + 255) / 256);

    // layer 1
    zero_kernel  <<<2048, 256, 0, stream>>>(agg, (long)NN * DF);
    zero_kernel  <<<256,  256, 0, stream>>>(dinv, (long)NN);
    degree_kernel<<<(EE + 255) / 256, 256, 0, stream>>>(dst, dinv);
    recip_kernel <<<(NN + 255) / 256, 256, 0, stream>>>(dinv);
    scatter_kernel<<<scatter_blocks, 256, 0, stream>>>(x, src, dst, agg);
    gemm1_kernel <<<(NN + 31) / 32, 128, 0, stream>>>(x, agg, dinv, W1l, W1r, b1, hbuf);

    // layer 2 (reuse agg buffer)
    zero_kernel  <<<2048, 256, 0, stream>>>(agg, (long)NN * DF);
    scatter_kernel<<<scatter_blocks, 256, 0, stream>>>(hbuf, src, dst, agg);
    gemm2_kernel <<<NN / 16, 32, 0, stream>>>(hbuf, agg, dinv, W2l, W2r, b2, out);
}